// GTvelo_58935541235939
// MI455X (gfx1250) — compile-verified
//
#include <hip/hip_runtime.h>
#include <math.h>

#define NNODES 50000
#define NEDGES 800000
#define INF_   50
#define HID_   64
#define HEADS_ 4
#define CH_    16
#define NC_    12
#define INV_SQRT_C 0.25f
#define LN_EPS 1e-5f

typedef float    v2f  __attribute__((ext_vector_type(2)));
typedef float    v8f  __attribute__((ext_vector_type(8)));

// codegen-confirmed on this toolchain (round-1 asm shows v_wmma_f32_16x16x4_f32)
#define WMMA4(a, b, c) \
  __builtin_amdgcn_wmma_f32_16x16x4_f32(false, (a), false, (b), (short)0, (c), false, false)

// ---------------------------------------------------------------------------
// Generic WMMA GEMM: out[M x NCOL] = A[M x K] @ W[K x NCOL] + bias, opt ReLU.
// K, NCOL compile-time so bounds logic constant-folds; remaining edge cases
// (K=50 tail, NCOL=12) use clamped addresses + v_cndmask (no exec branching).
// Fragment layouts per CDNA5 ISA 7.12.2:
//   A-frag (f32 16x4): lane l holds row (l&15), K = k0 + 2*(l>>4) + {0,1}
//   B-frag (f32 4x16): lane l holds col (l&15), same K pattern
//   D (16x16 f32):     vgpr r, lane l -> row r + 8*(l>>4), col (l&15)
// ---------------------------------------------------------------------------
template <int K, int NCOL, bool RELU>
__global__ void gemm_wmma(const float* __restrict__ A,
                          const float* __restrict__ W,
                          const float* __restrict__ bias,
                          float* __restrict__ out, int M) {
  const int lane = threadIdx.x & 31;
  const int wv   = threadIdx.x >> 5;
  const int lid  = lane & 15;
  const int half = lane >> 4;
  const int rowBase = blockIdx.x * 16;
  const int colBase = wv * 16;

  int arow = rowBase + lid;
  if (arow >= M) arow = M - 1;                 // clamp; OOB rows never stored
  const int col = colBase + lid;
  constexpr bool CG = (NCOL & 15) != 0;        // column guard needed?
  const bool colOK = CG ? (col < NCOL) : true;
  const int coln   = CG ? (colOK ? col : NCOL - 1) : col;

  v8f c = {};
  constexpr int KSTEPS = (K + 3) / 4;
#pragma unroll
  for (int s = 0; s < KSTEPS; ++s) {
    const int k0 = s * 4 + half * 2;
    int kc = k0;
    if (K & 3) kc = (k0 <= K - 2) ? k0 : K - 2;  // clamp (K even); folds for K%4==0
    v2f a, b;
    a.x = A[arow * K + kc];
    a.y = A[arow * K + kc + 1];
    b.x = W[(kc)     * NCOL + coln];
    b.y = W[(kc + 1) * NCOL + coln];
    if ((K & 3) && kc != k0) { a.x = 0.f; a.y = 0.f; b.x = 0.f; b.y = 0.f; }
    if (CG && !colOK)        { b.x = 0.f; b.y = 0.f; }
    c = WMMA4(a, b, c);
  }

  const float bvv = colOK ? bias[coln] : 0.f;
#pragma unroll
  for (int r = 0; r < 8; ++r) {
    const int row = rowBase + r + half * 8;
    if (row < M && colOK) {
      float val = c[r] + bvv;
      if (RELU) val = val > 0.f ? val : 0.f;
      out[row * NCOL + col] = val;
    }
  }
}

// ---------------------------------------------------------------------------
// Fused Q/K/V/Skip GEMM: reads each A fragment once, 4 WMMAs per k-step.
// K = NCOL = 64, no guards at all.
// ---------------------------------------------------------------------------
__global__ void gemm_qkvs(const float* __restrict__ A,
                          const float* __restrict__ Wq_, const float* __restrict__ Wk_,
                          const float* __restrict__ Wv_, const float* __restrict__ Ws_,
                          const float* __restrict__ bq_, const float* __restrict__ bk_,
                          const float* __restrict__ bv_, const float* __restrict__ bs_,
                          float* __restrict__ oq, float* __restrict__ ok_,
                          float* __restrict__ ov, float* __restrict__ os,
                          int M) {
  const int lane = threadIdx.x & 31;
  const int wv   = threadIdx.x >> 5;
  const int lid  = lane & 15;
  const int half = lane >> 4;
  const int rowBase = blockIdx.x * 16;
  const int col     = wv * 16 + lid;

  int arow = rowBase + lid;
  if (arow >= M) arow = M - 1;

  v8f cq = {}, ck = {}, cv = {}, cs = {};
#pragma unroll
  for (int s = 0; s < 16; ++s) {
    const int k0 = s * 4 + half * 2;
    v2f a, b;
    a.x = A[arow * 64 + k0];
    a.y = A[arow * 64 + k0 + 1];
    b.x = Wq_[k0 * 64 + col]; b.y = Wq_[(k0 + 1) * 64 + col]; cq = WMMA4(a, b, cq);
    b.x = Wk_[k0 * 64 + col]; b.y = Wk_[(k0 + 1) * 64 + col]; ck = WMMA4(a, b, ck);
    b.x = Wv_[k0 * 64 + col]; b.y = Wv_[(k0 + 1) * 64 + col]; cv = WMMA4(a, b, cv);
    b.x = Ws_[k0 * 64 + col]; b.y = Ws_[(k0 + 1) * 64 + col]; cs = WMMA4(a, b, cs);
  }

  const float vbq = bq_[col], vbk = bk_[col], vbv = bv_[col], vbs = bs_[col];
#pragma unroll
  for (int r = 0; r < 8; ++r) {
    const int row = rowBase + r + half * 8;
    if (row < M) {
      oq [row * 64 + col] = cq[r] + vbq;
      ok_[row * 64 + col] = ck[r] + vbk;
      ov [row * 64 + col] = cv[r] + vbv;
      os [row * 64 + col] = cs[r] + vbs;
    }
  }
}

// ---------------------------------------------------------------------------
// float atomic max via int punning (valid with 0xFFFFFFFF init pattern)
// ---------------------------------------------------------------------------
__device__ __forceinline__ void atomicMaxF(float* addr, float v) {
  if (v >= 0.f)
    atomicMax((int*)addr, __float_as_int(v));
  else
    atomicMin((unsigned int*)addr, __float_as_uint(v));
}

// pass 1: alpha per (edge, head) + segment max over dst
__global__ void edge_alpha_max(const int* __restrict__ src, const int* __restrict__ dst,
                               const float* __restrict__ ea,
                               const float* __restrict__ q, const float* __restrict__ k,
                               const float* __restrict__ we,
                               float* __restrict__ alphaB, float* __restrict__ amax,
                               int ne) {
  const int idx = blockIdx.x * blockDim.x + threadIdx.x;
  if (idx >= ne * HEADS_) return;
  const int e = idx >> 2, hd = idx & 3;
  const int s = src[e], d = dst[e];
  const float eav = ea[e];
  const float4* qd = (const float4*)(q + (size_t)d * HID_ + hd * CH_);
  const float4* ks = (const float4*)(k + (size_t)s * HID_ + hd * CH_);
  const float4* w  = (const float4*)(we + hd * CH_);
  float acc = 0.f;
#pragma unroll
  for (int c4 = 0; c4 < 4; ++c4) {
    const float4 qv = qd[c4], kv = ks[c4], wv = w[c4];
    acc += qv.x * (kv.x + eav * wv.x) + qv.y * (kv.y + eav * wv.y)
         + qv.z * (kv.z + eav * wv.z) + qv.w * (kv.w + eav * wv.w);
  }
  acc *= INV_SQRT_C;
  alphaB[idx] = acc;
  atomicMaxF(amax + d * HEADS_ + hd, acc);
}

// pass 2: exp(alpha - amax[dst]) in place + segment-sum denominator
__global__ void edge_exp_denom(const int* __restrict__ dst,
                               float* __restrict__ alphaB,
                               const float* __restrict__ amax,
                               float* __restrict__ denom, int ne) {
  const int idx = blockIdx.x * blockDim.x + threadIdx.x;
  if (idx >= ne * HEADS_) return;
  const int e = idx >> 2, hd = idx & 3;
  const int d = dst[e];
  const float ex = expf(alphaB[idx] - amax[d * HEADS_ + hd]);
  alphaB[idx] = ex;
  atomicAdd(denom + d * HEADS_ + hd, ex);
}

// pass 3: weighted message scatter-add
__global__ void edge_scatter(const int* __restrict__ src, const int* __restrict__ dst,
                             const float* __restrict__ ea,
                             const float* __restrict__ v, const float* __restrict__ we,
                             const float* __restrict__ alphaB,
                             const float* __restrict__ denom,
                             float* __restrict__ outb, int ne) {
  const int idx = blockIdx.x * blockDim.x + threadIdx.x;
  if (idx >= ne * HEADS_) return;
  const int e = idx >> 2, hd = idx & 3;
  const int s = src[e], d = dst[e];
  const float p = alphaB[idx] / (denom[d * HEADS_ + hd] + 1e-16f);
  const float eav = ea[e];
  const float4* vs = (const float4*)(v + (size_t)s * HID_ + hd * CH_);
  const float4* w  = (const float4*)(we + hd * CH_);
  float* ob = outb + (size_t)d * HID_ + hd * CH_;
#pragma unroll
  for (int c4 = 0; c4 < 4; ++c4) {
    const float4 vv = vs[c4], wv = w[c4];
    atomicAdd(ob + c4 * 4 + 0, (vv.x + eav * wv.x) * p);
    atomicAdd(ob + c4 * 4 + 1, (vv.y + eav * wv.y) * p);
    atomicAdd(ob + c4 * 4 + 2, (vv.z + eav * wv.z) * p);
    atomicAdd(ob + c4 * 4 + 3, (vv.w + eav * wv.w) * p);
  }
}

// beta gate + residual + layernorm, fused, one wave32 per node (2 feats/lane)
__global__ void node_gate_ln(float* __restrict__ h, const float* __restrict__ outb,
                             const float* __restrict__ xr,
                             const float* __restrict__ wbeta,
                             const float* __restrict__ g, const float* __restrict__ b,
                             int n) {
  const int lane = threadIdx.x & 31;
  const int node = blockIdx.x * (blockDim.x >> 5) + (threadIdx.x >> 5);
  if (node >= n) return;
  const int base = node * HID_;
  const float o0 = outb[base + lane],  o1 = outb[base + 32 + lane];
  const float x0 = xr[base + lane],    x1 = xr[base + 32 + lane];
  float part = o0 * wbeta[lane]              + o1 * wbeta[32 + lane]
             + x0 * wbeta[64 + lane]         + x1 * wbeta[96 + lane]
             + (o0 - x0) * wbeta[128 + lane] + (o1 - x1) * wbeta[160 + lane];
#pragma unroll
  for (int off = 16; off; off >>= 1) part += __shfl_xor(part, off, 32);
  const float beta = 1.f / (1.f + expf(-part));
  const float g0 = beta * x0 + (1.f - beta) * o0;
  const float g1 = beta * x1 + (1.f - beta) * o1;
  const float z0 = g0 + h[base + lane];
  const float z1 = g1 + h[base + 32 + lane];
  float sm = z0 + z1;
#pragma unroll
  for (int off = 16; off; off >>= 1) sm += __shfl_xor(sm, off, 32);
  const float mu = sm * (1.f / 64.f);
  const float d0 = z0 - mu, d1 = z1 - mu;
  float vs = d0 * d0 + d1 * d1;
#pragma unroll
  for (int off = 16; off; off >>= 1) vs += __shfl_xor(vs, off, 32);
  const float inv = 1.f / sqrtf(vs * (1.f / 64.f) + LN_EPS);
  h[base + lane]      = d0 * inv * g[lane]      + b[lane];
  h[base + 32 + lane] = d1 * inv * g[32 + lane] + b[32 + lane];
}

extern "C" void kernel_launch(void* const* d_in, const int* in_sizes, int n_in,
                              void* d_out, int out_size, void* d_ws, size_t ws_size,
                              hipStream_t stream) {
  const float* x     = (const float*)d_in[0];
  const int*   ei    = (const int*)d_in[1];
  const float* ea    = (const float*)d_in[2];
  const float* Win   = (const float*)d_in[3];
  const float* b_in  = (const float*)d_in[4];
  const float* Wq    = (const float*)d_in[5];
  const float* bq    = (const float*)d_in[6];
  const float* Wk    = (const float*)d_in[7];
  const float* bk    = (const float*)d_in[8];
  const float* Wv    = (const float*)d_in[9];
  const float* bv    = (const float*)d_in[10];
  const float* Wedge = (const float*)d_in[11];
  const float* Wskip = (const float*)d_in[12];
  const float* bskip = (const float*)d_in[13];
  const float* Wbeta = (const float*)d_in[14];
  const float* ln_g  = (const float*)d_in[15];
  const float* ln_b  = (const float*)d_in[16];
  const float* Wc1   = (const float*)d_in[17];
  const float* bc1   = (const float*)d_in[18];
  const float* Wc2   = (const float*)d_in[19];
  const float* bc2   = (const float*)d_in[20];
  float* out = (float*)d_out;

  const int n = NNODES, e = NEDGES;
  const int* src = ei;        // edge_index[0]
  const int* dst = ei + e;    // edge_index[1]

  float* ws     = (float*)d_ws;
  float* h      = ws;                       // n*64
  float* qb     = h    + (size_t)n * 64;    // n*64
  float* kb     = qb   + (size_t)n * 64;    // n*64
  float* vb     = kb   + (size_t)n * 64;    // n*64
  float* xr     = vb   + (size_t)n * 64;    // n*64
  float* outb   = xr   + (size_t)n * 64;    // n*64
  float* alphaB = outb + (size_t)n * 64;    // e*4
  float* amax   = alphaB + (size_t)e * 4;   // n*4
  float* denom  = amax + (size_t)n * 4;     // n*4
  float* h1     = qb;                       // reuse for classifier hidden

  dim3 gM((n + 15) / 16);
  dim3 gE((e * HEADS_ + 255) / 256);
  dim3 gN((n + 7) / 8);

  // h = x @ Win + b_in  (K=50, tail handled branch-free)
  gemm_wmma<INF_, HID_, false><<<gM, 128, 0, stream>>>(x, Win, b_in, h, n);

  for (int i = 0; i < 3; ++i) {
    const float* we = Wedge + i * HID_;
    gemm_qkvs<<<gM, 128, 0, stream>>>(h,
        Wq + i * 4096, Wk + i * 4096, Wv + i * 4096, Wskip + i * 4096,
        bq + i * 64,   bk + i * 64,   bv + i * 64,   bskip + i * 64,
        qb, kb, vb, xr, n);

    hipMemsetAsync(outb, 0, (size_t)n * 64 * sizeof(float), stream);
    hipMemsetAsync(denom, 0, (size_t)n * 4 * sizeof(float), stream);
    hipMemsetAsync(amax, 0xFF, (size_t)n * 4 * sizeof(float), stream);

    edge_alpha_max<<<gE, 256, 0, stream>>>(src, dst, ea, qb, kb, we, alphaB, amax, e);
    edge_exp_denom<<<gE, 256, 0, stream>>>(dst, alphaB, amax, denom, e);
    edge_scatter<<<gE, 256, 0, stream>>>(src, dst, ea, vb, we, alphaB, denom, outb, e);

    node_gate_ln<<<gN, 256, 0, stream>>>(h, outb, xr, Wbeta + i * 192,
                                         ln_g + i * 64, ln_b + i * 64, n);
  }

  // classifier
  gemm_wmma<HID_, HID_, true><<<gM, 128, 0, stream>>>(h, Wc1, bc1, h1, n);
  gemm_wmma<HID_, NC_, false><<<gM, 32, 0, stream>>>(h1, Wc2, bc2, out, n);
}